// GEMLayer_16758962389084
// MI455X (gfx1250) — compile-verified
//
#include <hip/hip_runtime.h>
#include <hip/hip_bf16.h>

// out = relu(x @ W + (sum_d A_d @ h) @ V)
//   - softmax over size-1 axis == 1.0  -> alpha unused
//   - shared V factors out of the per-device loop -> single scatter + single GEMM
//
// Stage 1: zero S[N,128] in workspace
// Stage 2: edge scatter  S[r] += val * h[c]   (one wave32 per edge, f32 HW atomics)
// Stage 3: fused WMMA GEMM  out = relu(x@W + S@V)  using V_WMMA_F32_16X16X4_F32

#define DOUT 128
#define DIN  256

typedef float v2f __attribute__((ext_vector_type(2)));
typedef float v8f __attribute__((ext_vector_type(8)));

#if defined(__gfx1250__) && !__has_builtin(__builtin_amdgcn_wmma_f32_16x16x4_f32)
#error "gfx1250: __builtin_amdgcn_wmma_f32_16x16x4_f32 not available"
#endif

// ---------------------------------------------------------------- Stage 1
__global__ void gem_zero_f4(float4* __restrict__ p, long n4) {
    long i = (long)blockIdx.x * blockDim.x + threadIdx.x;
    long s = (long)gridDim.x * blockDim.x;
    for (; i < n4; i += s) p[i] = make_float4(0.f, 0.f, 0.f, 0.f);
}

// ---------------------------------------------------------------- Stage 2
// One wave per edge: lane L handles columns [4L, 4L+4) of the 128-wide row.
__global__ void gem_edge_scatter(const int* __restrict__ rows,
                                 const int* __restrict__ cols,
                                 const float* __restrict__ vals,
                                 const float* __restrict__ h,
                                 float* __restrict__ S, long nE) {
    const int  lane = threadIdx.x & 31;
    long wid = (long)blockIdx.x * (blockDim.x >> 5) + (threadIdx.x >> 5);
    long wst = (long)gridDim.x * (blockDim.x >> 5);
    for (long e = wid; e < nE; e += wst) {
        const int   r = rows[e];
        const int   c = cols[e];
        const float v = vals[e];
        float4 hv = ((const float4*)(h + (long)c * DOUT))[lane];
        float* dst = S + (long)r * DOUT + lane * 4;
        __hip_atomic_fetch_add(dst + 0, v * hv.x, __ATOMIC_RELAXED, __HIP_MEMORY_SCOPE_AGENT);
        __hip_atomic_fetch_add(dst + 1, v * hv.y, __ATOMIC_RELAXED, __HIP_MEMORY_SCOPE_AGENT);
        __hip_atomic_fetch_add(dst + 2, v * hv.z, __ATOMIC_RELAXED, __HIP_MEMORY_SCOPE_AGENT);
        __hip_atomic_fetch_add(dst + 3, v * hv.w, __ATOMIC_RELAXED, __HIP_MEMORY_SCOPE_AGENT);
    }
}

// ---------------------------------------------------------------- Stage 3
// One wave computes a 16-row x 128-col output strip: 8 accumulator tiles of
// 16x16, fed by V_WMMA_F32_16X16X4_F32 (K=4 per op).
//
// f32 WMMA operand layouts (ISA 7.12.2, wave32):
//   A 16x4:  lane l holds (A[l&15][2*(l>>4)], A[l&15][2*(l>>4)+1]) in a v2f
//   B 4x16:  lane l holds (B[2*(l>>4)][l&15], B[2*(l>>4)+1][l&15]) in a v2f
//   C/D 16x16: 8 VGPRs; elem i of v8f = D[(l>>4)*8 + i][l&15]
__global__ void __launch_bounds__(256)
gem_fused_wmma(const float* __restrict__ x,  const float* __restrict__ W,
               const float* __restrict__ S,  const float* __restrict__ V,
               float* __restrict__ out, int N) {
    const int lane    = threadIdx.x & 31;
    const int waveBlk = threadIdx.x >> 5;
    const int rowTile = blockIdx.x * (blockDim.x >> 5) + waveBlk;
    const int r0      = rowTile * 16;
    if (r0 >= N) return;                    // wave-uniform: EXEC stays all-1s

    const int m  = lane & 15;               // row within A tile / col within B tile
    const int kb = (lane >> 4) * 2;         // K sub-offset: 0 or 2

    v8f acc[8] = {};                        // 8 tiles x 8 VGPRs = 64 VGPRs

    // ---- x @ W, K = 256 ----
    const float* xrow = x + (long)(r0 + m) * DIN;
    for (int k = 0; k < DIN; k += 4) {
        v2f a = *(const v2f*)(xrow + k + kb);     // 8B-aligned (k%4==0, kb even)
        const float* w0 = W + (long)(k + kb)     * DOUT;
        const float* w1 = W + (long)(k + kb + 1) * DOUT;
#pragma unroll
        for (int t = 0; t < 8; ++t) {
            const int c = t * 16 + m;
            v2f b; b.x = w0[c]; b.y = w1[c];
            acc[t] = __builtin_amdgcn_wmma_f32_16x16x4_f32(
                false, a, false, b, (short)0, acc[t], false, false);
        }
    }

    // ---- S @ V, K = 128, accumulate into the same tiles ----
    const float* srow = S + (long)(r0 + m) * DOUT;
    for (int k = 0; k < DOUT; k += 4) {
        v2f a = *(const v2f*)(srow + k + kb);
        const float* v0 = V + (long)(k + kb)     * DOUT;
        const float* v1 = V + (long)(k + kb + 1) * DOUT;
#pragma unroll
        for (int t = 0; t < 8; ++t) {
            const int c = t * 16 + m;
            v2f b; b.x = v0[c]; b.y = v1[c];
            acc[t] = __builtin_amdgcn_wmma_f32_16x16x4_f32(
                false, a, false, b, (short)0, acc[t], false, false);
        }
    }

    // ---- ReLU + store:  D elem i -> row (lane>>4)*8 + i, col t*16 + (lane&15)
    const int rbase = r0 + (lane >> 4) * 8;
#pragma unroll
    for (int t = 0; t < 8; ++t) {
        const int c = t * 16 + (lane & 15);
#pragma unroll
        for (int i = 0; i < 8; ++i) {
            float vv = acc[t][i];
            out[(long)(rbase + i) * DOUT + c] = vv > 0.f ? vv : 0.f;
        }
    }
}

// ---------------------------------------------------------------- launch
extern "C" void kernel_launch(void* const* d_in, const int* in_sizes, int n_in,
                              void* d_out, int out_size, void* d_ws, size_t ws_size,
                              hipStream_t stream) {
    const float* x    = (const float*)d_in[0];
    const float* h    = (const float*)d_in[1];
    const int*   rows = (const int*)  d_in[2];
    const int*   cols = (const int*)  d_in[3];
    const float* vals = (const float*)d_in[4];
    const float* W    = (const float*)d_in[5];
    const float* V    = (const float*)d_in[6];
    // d_in[7] (alpha): softmax over a size-1 axis is identically 1 -> unused.

    const int  N  = in_sizes[1] / DOUT;       // 50000
    const long nE = (long)in_sizes[2];        // D*E = 6.4M edges
    float* S = (float*)d_ws;                  // N*128 floats = 25.6 MB scratch

    // Stage 1: zero accumulator
    gem_zero_f4<<<1024, 256, 0, stream>>>((float4*)S, (long)N * DOUT / 4);

    // Stage 2: scatter-accumulate all D devices' edges into S
    gem_edge_scatter<<<4096, 256, 0, stream>>>(rows, cols, vals, h, S, nE);

    // Stage 3: fused WMMA GEMM + ReLU
    const int rowTiles = (N + 15) / 16;       // 3125
    const int wavesPerBlock = 256 / 32;       // 8
    const int blocks = (rowTiles + wavesPerBlock - 1) / wavesPerBlock;
    gem_fused_wmma<<<blocks, 256, 0, stream>>>(x, W, S, V, (float*)d_out, N);
}